// ManifoldModel_62526133895556
// MI455X (gfx1250) — compile-verified
//
#include <hip/hip_runtime.h>
#include <math.h>

// ---------------------------------------------------------------------------
// Problem constants (from the reference)
// ---------------------------------------------------------------------------
#define B_ROWS 4096
#define KPAIR  4
#define DDIM   512
#define FEATD  2048
#define RREL   9
#define AACT   7
#define NCOMBO (RREL * AACT)   // 63
#define MARGIN 0.5f
#define EPSV   1e-6f

typedef float v2f __attribute__((ext_vector_type(2)));
typedef float v8f __attribute__((ext_vector_type(8)));
typedef unsigned int u32x4 __attribute__((ext_vector_type(4)));
typedef int i32x8 __attribute__((ext_vector_type(8)));
typedef int i32x4 __attribute__((ext_vector_type(4)));

// ---------------------------------------------------------------------------
// Tensor Data Mover: load one 16-row x 128-col f32 tile of `img` into LDS,
// with 4-dword padding after every 128 dwords (row stride 132 in LDS ->
// conflict-free ds_load_b64 A-fragment reads).
// D# layout per CDNA5 ISA 8.3/8.4 (group0 128b, group1 256b; groups 2/3 zero
// => 2D tensor). This toolchain exposes the 6-arg builtin form.
// ---------------------------------------------------------------------------
__device__ __forceinline__ void tdm_load_a_tile(const float* gaddr, unsigned lds_off) {
    const unsigned long long ga = (unsigned long long)(uintptr_t)gaddr;
    u32x4 g0;
    g0.x = 1u;                                               // count=1 (valid user D#)
    g0.y = lds_off;                                          // lds_addr (bytes)
    g0.z = (unsigned)ga;                                     // global_addr[31:0]
    g0.w = ((unsigned)(ga >> 32) & 0x1FFFFFFu) | (2u << 30); // global_addr[56:32] | type=2

    i32x8 g1;
    g1[0] = (2 << 16)            // data_size = 4 bytes
          | (1 << 20)            // pad_enable
          | (6 << 22)            // pad_interval: every 128 dwords
          | (3 << 25);           // pad_amount: 4 dwords
    g1[1] = (int)(((unsigned)FEATD & 0xFFFFu) << 16);        // tensor_dim0[15:0] @bit48
    g1[2] = (int)(((unsigned)B_ROWS & 0xFFFFu) << 16);       // dim0 hi=0 | tensor_dim1[15:0]
    g1[3] = (int)(128u << 16);                               // dim1 hi=0 | tile_dim0=128
    g1[4] = 16;                                              // tile_dim1=16, tile_dim2=0
    g1[5] = FEATD;                                           // tensor_dim0_stride[31:0]
    g1[6] = 0;                                               // stride hi | dim1_stride
    g1[7] = 0;
    const i32x4 z4 = {0, 0, 0, 0};
    const i32x8 z8 = {0, 0, 0, 0, 0, 0, 0, 0};
    __builtin_amdgcn_tensor_load_to_lds(g0, g1, z4, z4, z8, 0);
}

// ---------------------------------------------------------------------------
// Kernel 1: combo table  T[a*9+r][i] = relu( sum_j action_ops[r,i,j]*obj_emb[a,j] )
// ---------------------------------------------------------------------------
__global__ void __launch_bounds__(256)
build_table_kernel(const float* __restrict__ obj_emb,
                   const float* __restrict__ action_ops,
                   float* __restrict__ table) {
    __shared__ float sobj[DDIM];
    const int c   = blockIdx.x;          // combo id = a*9 + r
    const int a   = c / RREL;
    const int r   = c % RREL;
    const int tid = threadIdx.x;

    sobj[tid]       = obj_emb[a * DDIM + tid];
    sobj[tid + 256] = obj_emb[a * DDIM + tid + 256];
    __syncthreads();

    const float* opbase = action_ops + (size_t)r * DDIM * DDIM;
    for (int i = tid; i < DDIM; i += 256) {
        const float4* rowp = (const float4*)(opbase + (size_t)i * DDIM);
        float acc = 0.f;
#pragma unroll 8
        for (int j4 = 0; j4 < DDIM / 4; ++j4) {
            float4 w = rowp[j4];
            acc += w.x * sobj[j4 * 4 + 0];
            acc += w.y * sobj[j4 * 4 + 1];
            acc += w.z * sobj[j4 * 4 + 2];
            acc += w.w * sobj[j4 * 4 + 3];
        }
        table[c * DDIM + i] = fmaxf(acc, 0.f);
    }
}

// ---------------------------------------------------------------------------
// Kernel 2: fused  feature = relu(img@W + b)  via V_WMMA_F32_16X16X4_F32 with
// TDM double-buffered A staging, then triplet distances vs table gathers.
// Block = 256 threads (8 waves) -> 16 rows x 512 cols.
// ---------------------------------------------------------------------------
__global__ void __launch_bounds__(256)
fused_gemm_triplet_kernel(const float* __restrict__ img,
                          const float* __restrict__ Wm,
                          const float* __restrict__ bias,
                          const int*   __restrict__ actor_idx,
                          const int*   __restrict__ action_idx,
                          const int*   __restrict__ counts,
                          const int*   __restrict__ neg_actor,
                          const int*   __restrict__ neg_action,
                          const float* __restrict__ table,
                          float*       __restrict__ hinge) {
    constexpr int KC     = 128;          // K-chunk per TDM tile
    constexpr int NCHUNK = FEATD / KC;   // 16
    __shared__ __align__(16) float lds_a[2][16][132]; // double-buffered, padded rows
    __shared__ float red_dp[256][8];
    __shared__ float red_dn[256][8];
    __shared__ float l2dp[8][16];
    __shared__ float l2dn[8][16];
    __shared__ int   sh_combo[16][KPAIR];
    __shared__ int   sh_cnt[16];
    __shared__ int   sh_negc[16];

    const int tid  = threadIdx.x;
    const int wave = tid >> 5;         // 0..7
    const int lane = tid & 31;         // wave32 lane
    const int hi   = (lane >> 4) & 1;  // half-wave select
    const int lm   = lane & 15;        // row (A/C) / col (B/C) within tile
    const int row0 = blockIdx.x * 16;

    if (tid < 16) {
        const int rr = row0 + tid;
        sh_cnt[tid]  = counts[rr];
        sh_negc[tid] = neg_actor[rr] * RREL + neg_action[rr];
#pragma unroll
        for (int k = 0; k < KPAIR; ++k)
            sh_combo[tid][k] = actor_idx[rr * KPAIR + k] * RREL + action_idx[rr * KPAIR + k];
    }

    v8f acc[4] = {};   // 4 tiles x 8 VGPRs, exact fp32 accumulation

    const unsigned lds0  = (unsigned)(uintptr_t)(&lds_a[0][0][0]);
    const unsigned lds1  = lds0 + 16u * 132u * 4u;
    const float*   gbase = img + (size_t)row0 * FEATD;

    // prologue: kick off chunk 0 DMA
    if (wave == 0) tdm_load_a_tile(gbase, lds0);

    for (int c = 0; c < NCHUNK; ++c) {
        if (wave == 0) __builtin_amdgcn_s_wait_tensorcnt(0); // chunk c landed in LDS
        __syncthreads();
        if (wave == 0 && (c + 1) < NCHUNK)                   // prefetch chunk c+1
            tdm_load_a_tile(gbase + (size_t)(c + 1) * KC, (c & 1) ? lds0 : lds1);

        const int   buf = c & 1;
        const int   kb  = c * KC;
        const float* lrow = &lds_a[buf][lm][0];

        for (int kk = 0; kk < KC / 4; ++kk) {
            // A frag (ISA 7.12.2, 32-bit A 16x4): lane<16 -> K{0,1}, lane>=16 -> K{2,3}
            const int kloc = kk * 4 + 2 * hi;
            v2f afrag;
            afrag.x = lrow[kloc];
            afrag.y = lrow[kloc + 1];
            const int kglob = kb + kloc;
#pragma unroll
            for (int t = 0; t < 4; ++t) {
                // B frag (4x16 f32): v0 = B[K=2*hi][n], v1 = B[K=2*hi+1][n], n = lane&15
                const int ncol = wave * 64 + t * 16 + lm;
                v2f bfrag;
                bfrag.x = Wm[(size_t)kglob * DDIM + ncol];
                bfrag.y = Wm[(size_t)(kglob + 1) * DDIM + ncol];
                acc[t] = __builtin_amdgcn_wmma_f32_16x16x4_f32(
                    false, afrag, false, bfrag, (short)0, acc[t], false, false);
            }
        }
    }
    __syncthreads();

    // ---- distance pass: f = relu(acc + b), diff vs positive/negative -------
    float dp_l[8], dn_l[8];
#pragma unroll
    for (int v = 0; v < 8; ++v) { dp_l[v] = 0.f; dn_l[v] = 0.f; }

#pragma unroll
    for (int t = 0; t < 4; ++t) {
        const int   ncol = wave * 64 + t * 16 + lm;
        const float bb   = bias[ncol];
#pragma unroll
        for (int v = 0; v < 8; ++v) {
            const int   row = v + 8 * hi;   // C/D layout: VGPR v holds row v (+8 for hi lanes)
            const float f   = fmaxf(acc[t][v] + bb, 0.f);
            const int   cnt = sh_cnt[row];
            float p = 0.f;
#pragma unroll
            for (int k = 0; k < KPAIR; ++k) {
                // branch-free: indices always valid, only the mask is ragged
                const float tv = table[sh_combo[row][k] * DDIM + ncol];
                p += ((k < cnt) ? 1.0f : 0.0f) * tv;
            }
            p /= (float)cnt;
            const float ng = table[sh_negc[row] * DDIM + ncol];
            const float d1 = f - p  + EPSV; dp_l[v] += d1 * d1;
            const float d2 = f - ng + EPSV; dn_l[v] += d2 * d2;
        }
    }

#pragma unroll
    for (int v = 0; v < 8; ++v) { red_dp[tid][v] = dp_l[v]; red_dn[tid][v] = dn_l[v]; }
    __syncthreads();

    // level-1: 128 threads, each folds 16 lanes of one (wave,row)
    if (tid < 128) {
        const int w2   = tid >> 4;
        const int row  = tid & 15;
        const int base = w2 * 32 + (row >= 8 ? 16 : 0);
        const int v    = row & 7;
        float sdp = 0.f, sdn = 0.f;
#pragma unroll
        for (int j = 0; j < 16; ++j) { sdp += red_dp[base + j][v]; sdn += red_dn[base + j][v]; }
        l2dp[w2][row] = sdp; l2dn[w2][row] = sdn;
    }
    __syncthreads();

    // level-2: 16 threads fold 8 waves, emit hinge per row
    if (tid < 16) {
        float sdp = 0.f, sdn = 0.f;
#pragma unroll
        for (int w = 0; w < 8; ++w) { sdp += l2dp[w][tid]; sdn += l2dn[w][tid]; }
        const float h = sqrtf(sdp) - sqrtf(sdn) + MARGIN;
        hinge[row0 + tid] = fmaxf(h, 0.f);
    }
}

// ---------------------------------------------------------------------------
// Kernel 3: deterministic mean of 4096 hinge values
// ---------------------------------------------------------------------------
__global__ void __launch_bounds__(256)
reduce_mean_kernel(const float* __restrict__ hinge, float* __restrict__ out) {
    __shared__ float sred[256];
    const int tid = threadIdx.x;
    float s = 0.f;
    for (int i = tid; i < B_ROWS; i += 256) s += hinge[i];
    sred[tid] = s;
    __syncthreads();
    for (int off = 128; off > 0; off >>= 1) {
        if (tid < off) sred[tid] += sred[tid + off];
        __syncthreads();
    }
    if (tid == 0) out[0] = sred[0] / (float)B_ROWS;
}

// ---------------------------------------------------------------------------
// Launcher
// ---------------------------------------------------------------------------
extern "C" void kernel_launch(void* const* d_in, const int* in_sizes, int n_in,
                              void* d_out, int out_size, void* d_ws, size_t ws_size,
                              hipStream_t stream) {
    (void)in_sizes; (void)n_in; (void)out_size; (void)ws_size;

    const float* img        = (const float*)d_in[0];  // [4096,2048]
    const float* Wm         = (const float*)d_in[1];  // [2048,512]
    const float* bias       = (const float*)d_in[2];  // [512]
    const float* obj_emb    = (const float*)d_in[3];  // [7,512]
    const float* action_ops = (const float*)d_in[4];  // [9,512,512]
    const int*   actor_idx  = (const int*)d_in[5];    // [4096,4]
    const int*   action_idx = (const int*)d_in[6];    // [4096,4]
    const int*   counts     = (const int*)d_in[7];    // [4096]
    const int*   neg_actor  = (const int*)d_in[8];    // [4096]
    const int*   neg_action = (const int*)d_in[9];    // [4096]

    // workspace: combo table (63*512 f32 = 126KB) then hinge[4096] at 128KB
    float* table = (float*)d_ws;
    float* hinge = (float*)((char*)d_ws + (128u << 10));

    hipLaunchKernelGGL(build_table_kernel, dim3(NCOMBO), dim3(256), 0, stream,
                       obj_emb, action_ops, table);

    hipLaunchKernelGGL(fused_gemm_triplet_kernel, dim3(B_ROWS / 16), dim3(256), 0, stream,
                       img, Wm, bias, actor_idx, action_idx, counts,
                       neg_actor, neg_action, table, hinge);

    hipLaunchKernelGGL(reduce_mean_kernel, dim3(1), dim3(256), 0, stream,
                       hinge, (float*)d_out);
}